// BaseConvolution_29729763623230
// MI455X (gfx1250) — compile-verified
//
#include <hip/hip_runtime.h>

typedef float v2f __attribute__((ext_vector_type(2)));
typedef float v8f __attribute__((ext_vector_type(8)));
typedef __attribute__((address_space(3))) void lds_void;

#define B_      8
#define NP_     4096
#define STRIDE_ 4
#define K_      32
#define NS_     (NP_ / STRIDE_)     // 1024
#define NQ_     (B_ * NS_)          // 8192
#define DH_     128
#define DO_     256
#define QPW_    16                  // queries per MLP workgroup
#define KNNQ_   4                   // queries (waves) per KNN workgroup

// LDS byte offset of a __shared__ object (generic -> as3 -> 32-bit offset)
static __device__ __forceinline__ unsigned to_lds(const void* p) {
  return (unsigned)(size_t)(lds_void*)(void*)(const_cast<void*>(p));
}

// ---------------------------------------------------------------------------
// Kernel 1: exact KNN, one wave32 per query, zero barriers.
// Wave-private 4096-float distance row in LDS; argmin via shfl_xor butterfly;
// per-lane cached local best so only the winning lane rescans each pass.
// ---------------------------------------------------------------------------
__global__ __launch_bounds__(128) void knn_kernel(const float* __restrict__ pos,
                                                  int* __restrict__ nbr,
                                                  float* __restrict__ out_pos,
                                                  int* __restrict__ out_batch,
                                                  int* __restrict__ out_idx) {
  __shared__ float d2[KNNQ_][NP_];            // 64 KB, one row per wave
  const int tid  = threadIdx.x;
  const int wave = tid >> 5;
  const int lane = tid & 31;

  const int g = blockIdx.x * KNNQ_ + wave;    // query id
  const int b = g / NS_;
  const int s = g - b * NS_;
  const int q = b * NP_ + s * STRIDE_;

  float* drow = d2[wave];
  const float* pb = pos + (size_t)b * NP_ * 3;
  const float qx = pos[q * 3 + 0];
  const float qy = pos[q * 3 + 1];
  const float qz = pos[q * 3 + 2];

  // distances: lane handles j = lane + 32*t (coalesced pos reads, conflict-free LDS)
  for (int t = 0; t < NP_ / 32; ++t) {
    int j = lane + 32 * t;
    float dx = pb[j * 3 + 0] - qx;
    float dy = pb[j * 3 + 1] - qy;
    float dz = pb[j * 3 + 2] - qz;
    drow[j] = dx * dx + dy * dy + dz * dz;
  }
  if (lane < 3)  out_pos[g * 3 + lane] = pos[q * 3 + lane];
  if (lane == 3) { out_batch[g] = b; out_idx[g] = q; }

  // lane-local best over its private 128 slots (j strictly increasing -> low-idx tie)
  float bval = 3.0e38f; int bidx = 1 << 30;
  for (int t = 0; t < NP_ / 32; ++t) {
    int j = lane + 32 * t;
    float v = drow[j];
    if (v < bval) { bval = v; bidx = j; }
  }

  for (int k = 0; k < K_; ++k) {
    float rv = bval; int ri = bidx;
#pragma unroll
    for (int off = 16; off > 0; off >>= 1) {        // butterfly argmin, low-idx tie-break
      float ov = __shfl_xor(rv, off, 32);
      int   oi = __shfl_xor(ri, off, 32);
      if (ov < rv || (ov == rv && oi < ri)) { rv = ov; ri = oi; }
    }
    if (lane == 0) nbr[g * K_ + k] = b * NP_ + ri;
    if (bidx == ri) {                               // only the owner lane matches
      drow[ri] = 3.3e38f;
      bval = 3.0e38f; bidx = 1 << 30;
      for (int t = 0; t < NP_ / 32; ++t) {
        int j = lane + 32 * t;
        float v = drow[j];
        if (v < bval) { bval = v; bidx = j; }
      }
    }
  }
}

// ---------------------------------------------------------------------------
// Kernel 2: fused gather -> MLP1(67->128) -> ReLU -> MLP2(128->256) -> ReLU
//           -> max over K. 16 waves, 16 queries per WG.
// W2 B-fragments live in registers for the whole WG (query-invariant);
// W1 staged once in LDS; neighbor-feature gather uses async global->LDS.
// ---------------------------------------------------------------------------
__global__ __launch_bounds__(512) void mlp_kernel(const float* __restrict__ x,
                                                  const float* __restrict__ pos,
                                                  const float* __restrict__ W1,
                                                  const float* __restrict__ b1,
                                                  const float* __restrict__ W2,
                                                  const float* __restrict__ b2,
                                                  const int* __restrict__ nbr,
                                                  float* __restrict__ xout) {
  __shared__ float W1s[68 * 136];                  // 37 KB; stride 136 -> disjoint banks
  __shared__ __attribute__((aligned(16))) float feat[32 * 68];   // 8.5 KB
  __shared__ float    hbuf[32 * 132];              // 16.5 KB; stride 132
  __shared__ unsigned omax[DO_];
  __shared__ int      nidx[32];
  __shared__ float    psq[4];

  const int tid  = threadIdx.x;
  const int lane = tid & 31;
  const int wave = tid >> 5;                       // 0..15
  const int half = lane >> 4;                      // ISA 16x4 A/B K sub-slot
  const int lc   = lane & 15;

  // ---- one-time: W1 -> LDS (row 67 zero pad) ----
  for (int e = tid; e < 68 * DH_; e += 512) {
    int r = e >> 7, c = e & 127;
    W1s[r * 136 + c] = (r < 67) ? W1[r * DH_ + c] : 0.0f;
  }

  // ---- one-time: W2 B-fragments -> registers (query-invariant per wave) ----
  const int c2 = wave * 16 + lc;                   // GEMM2 output column
  v2f B2[32];
#pragma unroll
  for (int s2 = 0; s2 < 32; ++s2) {
    int ka = 4 * s2 + 2 * half;
    B2[s2].x = W2[ka * DO_ + c2];
    B2[s2].y = W2[(ka + 1) * DO_ + c2];
  }
  const float bias2 = b2[c2];

  const int n1 = wave & 7, m1 = wave >> 3;         // GEMM1 tile for this wave
  const int c1 = n1 * 16 + lc;
  const float bias1 = b1[c1];

  __syncthreads();

  for (int qi = 0; qi < QPW_; ++qi) {
    const int g = blockIdx.x * QPW_ + qi;
    if (tid < 32) nidx[tid] = nbr[g * 32 + tid];
    if (tid < 3) {
      int b = g / NS_, s = g - b * NS_;
      psq[tid] = pos[(b * NP_ + s * STRIDE_) * 3 + tid];
    }
    if (tid < DO_) omax[tid] = 0u;
    __syncthreads();

    // ---- gather: 32 rows x 64 floats == 512 float4 -> async global->LDS ----
    {
      int r = tid >> 4, c4 = tid & 15;
      const float* src = &x[(size_t)nidx[r] * 64 + c4 * 4];
      unsigned dst = to_lds(&feat[r * 68 + c4 * 4]);
      asm volatile("global_load_async_to_lds_b128 %0, %1, off"
                   :: "v"(dst), "v"(src) : "memory");
    }
    if (tid < 128) {                               // relative positions + zero pad col 67
      int r = tid >> 2, c = tid & 3;
      feat[r * 68 + 64 + c] =
          (c < 3) ? pos[(size_t)nidx[r] * 3 + c] - psq[c] : 0.0f;
    }
    asm volatile("s_wait_asynccnt 0x0" ::: "memory");
    __syncthreads();

    // ---- GEMM1: (32 x 68) @ (68 x 128); one 16x16 tile per wave ----
    {
      v8f acc = {0, 0, 0, 0, 0, 0, 0, 0};
#pragma unroll
      for (int k0 = 0; k0 < 68; k0 += 4) {
        int ka = k0 + 2 * half;
        v2f a = *(const v2f*)&feat[(m1 * 16 + lc) * 68 + ka];
        v2f bf;
        bf.x = W1s[ka * 136 + c1];
        bf.y = W1s[(ka + 1) * 136 + c1];
        acc = __builtin_amdgcn_wmma_f32_16x16x4_f32(false, a, false, bf,
                                                    (short)0, acc, false, false);
      }
#pragma unroll
      for (int i = 0; i < 8; ++i) {
        int r0 = m1 * 16 + half * 8 + i;           // C layout: VGPR i -> row half*8+i
        hbuf[r0 * 132 + c1] = fmaxf(acc[i] + bias1, 0.0f);
      }
    }
    __syncthreads();

    // ---- GEMM2: (32 x 128) @ (128 x 256); B from registers, A from LDS ----
    {
      v8f a20 = {0, 0, 0, 0, 0, 0, 0, 0};
      v8f a21 = {0, 0, 0, 0, 0, 0, 0, 0};
#pragma unroll
      for (int s2 = 0; s2 < 32; ++s2) {
        int ka = 4 * s2 + 2 * half;
        v2f a0 = *(const v2f*)&hbuf[lc * 132 + ka];
        v2f a1 = *(const v2f*)&hbuf[(16 + lc) * 132 + ka];
        a20 = __builtin_amdgcn_wmma_f32_16x16x4_f32(false, a0, false, B2[s2],
                                                    (short)0, a20, false, false);
        a21 = __builtin_amdgcn_wmma_f32_16x16x4_f32(false, a1, false, B2[s2],
                                                    (short)0, a21, false, false);
      }
      // bias + ReLU + max over this lane's 16 rows (start at 0 == ReLU fold)
      float mA = 0.0f;
#pragma unroll
      for (int i = 0; i < 8; ++i) {
        mA = fmaxf(mA, a20[i] + bias2);
        mA = fmaxf(mA, a21[i] + bias2);
      }
      atomicMax(&omax[c2], __float_as_uint(mA));   // >=0: uint order == float order
    }
    __syncthreads();

    if (tid < DO_) xout[(size_t)g * DO_ + tid] = __uint_as_float(omax[tid]);
    __syncthreads();
  }
}

// ---------------------------------------------------------------------------
extern "C" void kernel_launch(void* const* d_in, const int* in_sizes, int n_in,
                              void* d_out, int out_size, void* d_ws, size_t ws_size,
                              hipStream_t stream) {
  (void)in_sizes; (void)n_in; (void)out_size; (void)ws_size;
  const float* x   = (const float*)d_in[0];
  const float* pos = (const float*)d_in[1];
  // d_in[2] = batch (recomputed analytically)
  const float* W1  = (const float*)d_in[3];
  const float* b1  = (const float*)d_in[4];
  const float* W2  = (const float*)d_in[5];
  const float* b2  = (const float*)d_in[6];

  float* out       = (float*)d_out;
  float* xout      = out;                                   // 8192 x 256
  float* pos_out   = out + (size_t)NQ_ * DO_;               // 8192 x 3
  int*   batch_out = (int*)(out + (size_t)NQ_ * DO_ + (size_t)NQ_ * 3);
  int*   idx_out   = batch_out + NQ_;

  int* nbr = (int*)d_ws;                                    // 8192 x 32 ints

  knn_kernel<<<NQ_ / KNNQ_, 128, 0, stream>>>(pos, nbr, pos_out, batch_out, idx_out);
  mlp_kernel<<<NQ_ / QPW_, 512, 0, stream>>>(x, pos, W1, b1, W2, b2, nbr, xout);
}